// ImprovedOnsetOffsetLoss_83923660964093
// MI455X (gfx1250) — compile-verified
//
#include <hip/hip_runtime.h>

// ---------------------------------------------------------------------------
// ImprovedOnsetOffsetLoss for MI455X (gfx1250, wave32)
// 64 MB of f32 input @ 23.3 TB/s  ->  ~2.9 us memory-bound streaming pass.
// Deterministic tree reductions; cross-lane sums go through the matrix pipe
// via V_WMMA_F32_16X16X4_F32 (full f32 precision, B = ones).
// ---------------------------------------------------------------------------

typedef __attribute__((ext_vector_type(2))) float v2f;
typedef __attribute__((ext_vector_type(8))) float v8f;

#define B_ROWS 256
#define T_LEN 16384
#define SEGS 4
#define SEG_LEN (T_LEN / SEGS)            // 4096
#define NTHREADS 256                      // 8 waves (wave32)
#define PER_THREAD (SEG_LEN / NTHREADS)   // 16 elems/thread/array
#define NACC 9
#define WS_STRIDE 16                      // padded slot per block

__device__ __forceinline__ float sigmoidf_(float x) {
  return 1.0f / (1.0f + __expf(-x));
}
// logaddexp(0, x) = max(x,0) + log1p(exp(-|x|))
__device__ __forceinline__ float softplusf_(float x) {
  return fmaxf(x, 0.0f) + log1pf(__expf(-fabsf(x)));
}

// Full-precision f32 wave32 reduction through the WMMA pipe.
// A[m][0]=p(lane m), A[m][2]=p(lane 16+m), A[m][1]=A[m][3]=0; B = ones.
// D[m][n] = p_m + p_{m+16}; summing the 8 D VGPRs gives half-wave sums in
// lanes 0-15 / 16-31; one xor-16 shuffle completes the reduction.
__device__ __forceinline__ float wave_reduce_f32(float p) {
  v2f a; a[0] = p;    a[1] = 0.0f;
  v2f b; b[0] = 1.0f; b[1] = 1.0f;
  v8f c = {};
  c = __builtin_amdgcn_wmma_f32_16x16x4_f32(false, a, false, b,
                                            (short)0, c, false, false);
  float h = ((c[0] + c[1]) + (c[2] + c[3])) + ((c[4] + c[5]) + (c[6] + c[7]));
  h += __shfl_xor(h, 16, 32);
  return h;  // every lane holds the 32-lane sum
}

// acc indices: 0 on_pos_sum, 1 on_neg_sum, 2 on_pos_cnt,
//              3 off_pos_sum, 4 off_neg_sum, 5 off_pos_cnt,
//              6 sum(sigmoid(on)), 7 sum(sigmoid(off)), 8 consistency sum
__global__ void __launch_bounds__(NTHREADS)
loss_partial_kernel(const float* __restrict__ xon, const float* __restrict__ xoff,
                    const float* __restrict__ ton, const float* __restrict__ toff,
                    float* __restrict__ ws) {
  const int blk = blockIdx.x;           // 0 .. B_ROWS*SEGS-1
  const int row = blk / SEGS;
  const int seg = blk % SEGS;
  const int tid = threadIdx.x;
  const long rb = (long)row * T_LEN;
  const float* XON = xon + rb;
  const float* XOF = xoff + rb;
  const float* TON = ton + rb;
  const float* TOF = toff + rb;
  const int pos0 = seg * SEG_LEN + tid * PER_THREAD;

  float acc[NACC];
#pragma unroll
  for (int i = 0; i < NACC; ++i) acc[i] = 0.0f;

  // Prime the 3-deep onset-prob window (cached re-reads at chunk heads).
  float w0 = 0.0f, w1 = 0.0f, w2 = 0.0f;
  if (pos0 >= 3) {
    w0 = sigmoidf_(XON[pos0 - 3]);
    w1 = sigmoidf_(XON[pos0 - 2]);
    w2 = sigmoidf_(XON[pos0 - 1]);
  }

#pragma unroll
  for (int j = 0; j < PER_THREAD / 4; ++j) {
    const int t0 = pos0 + 4 * j;
    const float4 vxo = *reinterpret_cast<const float4*>(XON + t0);
    const float4 vxf = *reinterpret_cast<const float4*>(XOF + t0);
    const float4 vto = *reinterpret_cast<const float4*>(TON + t0);
    const float4 vtf = *reinterpret_cast<const float4*>(TOF + t0);
    const float xo[4] = {vxo.x, vxo.y, vxo.z, vxo.w};
    const float xf[4] = {vxf.x, vxf.y, vxf.z, vxf.w};
    const float to[4] = {vto.x, vto.y, vto.z, vto.w};
    const float tf[4] = {vtf.x, vtf.y, vtf.z, vtf.w};
#pragma unroll
    for (int k = 0; k < 4; ++k) {
      const int t = t0 + k;
      const float po = sigmoidf_(xo[k]);
      const float pf = sigmoidf_(xf[k]);
      const float lo = softplusf_(xo[k]) - xo[k] * to[k];
      const float lf = softplusf_(xf[k]) - xf[k] * tf[k];
      const bool pon = (to[k] == 1.0f);
      const bool pof = (tf[k] == 1.0f);
      acc[0] += pon ? fminf(lo, 5.0f) : 0.0f;   // MAX_POS clamp, positives
      acc[1] += pon ? 0.0f : lo;
      acc[2] += pon ? 1.0f : 0.0f;
      acc[3] += pof ? fminf(lf, 5.0f) : 0.0f;
      acc[4] += pof ? 0.0f : lf;
      acc[5] += pof ? 1.0f : 0.0f;
      acc[6] += po;
      acc[7] += pf;
      const float r = fmaxf(w0, fmaxf(w1, w2)); // max onset prob, t-1..t-3
      if (t >= 3 && r > 0.5f && pf > 0.5f) acc[8] += r * pf;
      w0 = w1; w1 = w2; w2 = po;
    }
  }

  // Wave-level WMMA reduction, then deterministic cross-wave LDS fold.
  __shared__ float sdata[NACC][NTHREADS / 32];
  const int lane = tid & 31;
  const int wave = tid >> 5;
#pragma unroll
  for (int m = 0; m < NACC; ++m) {
    const float s = wave_reduce_f32(acc[m]);
    if (lane == 0) sdata[m][wave] = s;
  }
  __syncthreads();
  if (tid < NACC) {
    float s = 0.0f;
#pragma unroll
    for (int w = 0; w < NTHREADS / 32; ++w) s += sdata[tid][w];
    ws[blk * WS_STRIDE + tid] = s;
  }
}

__global__ void __launch_bounds__(NTHREADS)
loss_final_kernel(const float* __restrict__ ws, float* __restrict__ out) {
  const int tid = threadIdx.x;          // == row index
  float r[NACC];
#pragma unroll
  for (int m = 0; m < NACC; ++m) r[m] = 0.0f;
#pragma unroll
  for (int s = 0; s < SEGS; ++s) {
    const float* p = ws + (size_t)(tid * SEGS + s) * WS_STRIDE;
#pragma unroll
    for (int m = 0; m < NACC; ++m) r[m] += p[m];
  }
  float v[NACC];
  v[0] = r[0]; v[1] = r[1]; v[2] = r[2];
  v[3] = r[3]; v[4] = r[4]; v[5] = r[5];
  v[6] = r[6];                          // global sum of onset probs
  v[7] = r[8];                          // global consistency sum
  v[8] = fabsf(r[6] - r[7]);            // per-row pairing term

  __shared__ float sdata[NACC][NTHREADS / 32];
  const int lane = tid & 31;
  const int wave = tid >> 5;
#pragma unroll
  for (int m = 0; m < NACC; ++m) {
    const float s = wave_reduce_f32(v[m]);
    if (lane == 0) sdata[m][wave] = s;
  }
  __syncthreads();

  if (tid == 0) {
    float S[NACC];
#pragma unroll
    for (int m = 0; m < NACC; ++m) {
      float s = 0.0f;
      for (int w = 0; w < NTHREADS / 32; ++w) s += sdata[m][w];
      S[m] = s;
    }
    const float N = (float)B_ROWS * (float)T_LEN;

    float pc = S[2], nc = N - pc;
    float pos_l = (pc > 0.0f) ? S[0] / fmaxf(pc, 1.0f) : 0.0f;
    float neg_l = (nc > 0.0f) ? S[1] / fmaxf(nc, 1.0f) : 0.0f;
    const float onset_loss = 1.5f * pos_l + neg_l;      // POS_COEFF = 1.5

    pc = S[5]; nc = N - pc;
    pos_l = (pc > 0.0f) ? S[3] / fmaxf(pc, 1.0f) : 0.0f;
    neg_l = (nc > 0.0f) ? S[4] / fmaxf(nc, 1.0f) : 0.0f;
    const float offset_loss = 1.5f * pos_l + neg_l;

    const float cons_loss = S[7] / ((float)B_ROWS * (float)(T_LEN - 3));
    const float pair_loss = S[8] / (float)B_ROWS;
    const float mean_on   = S[6] / N;
    const float d = mean_on - 0.0336f;                  // SPARSE_T
    const float sparse_loss = 0.0f * d * d;             // SPARSE_W = 0
    const float total = 5.0f * onset_loss + 3.0f * offset_loss +
                        0.1f * cons_loss + 0.05f * pair_loss + sparse_loss;

    out[0] = onset_loss;
    out[1] = offset_loss;
    out[2] = cons_loss;
    out[3] = pair_loss;
    out[4] = sparse_loss;
    out[5] = mean_on;
    out[6] = total;
  }
}

extern "C" void kernel_launch(void* const* d_in, const int* in_sizes, int n_in,
                              void* d_out, int out_size, void* d_ws, size_t ws_size,
                              hipStream_t stream) {
  (void)in_sizes; (void)n_in; (void)out_size; (void)ws_size;
  const float* xon  = (const float*)d_in[0];
  const float* xoff = (const float*)d_in[1];
  const float* ton  = (const float*)d_in[2];
  const float* toff = (const float*)d_in[3];
  float* ws  = (float*)d_ws;   // needs B_ROWS*SEGS*WS_STRIDE*4 = 64 KB
  float* out = (float*)d_out;  // 7 scalars

  loss_partial_kernel<<<B_ROWS * SEGS, NTHREADS, 0, stream>>>(xon, xoff, ton, toff, ws);
  loss_final_kernel<<<1, NTHREADS, 0, stream>>>(ws, out);
}